// ISWAPLayer_18726057410939
// MI455X (gfx1250) — compile-verified
//
#include <hip/hip_runtime.h>
#include <cstdint>

// iSWAP(t) on qubits (3,7) of a 16-qubit, batch-256 complex state.
// Memory-bound: 256 MB traffic, ~0.13 GFLOP -> AI ~= 0.5 flop/B, so the
// design target is pure bandwidth (B128, NT stores, gfx1250 async LDS DMA),
// not matrix throughput. All 128 MB of output is stored non-temporal so the
// 128 MB input stays resident in the 192 MB L2 across graph replays.

#define DIM_   65536
#define BATCH_ 256
#define ROWF4  64          // float4 per row (256 floats)
#define NBASE  16384       // indices with bits 12 and 8 cleared
#define BIT_A  (1u << 12)  // target qubit 3 (MSB-first ordering)
#define BIT_B  (1u << 8)   // target qubit 7

typedef __attribute__((ext_vector_type(4))) float v4f;

// Expand b in [0, 16384) to a 16-bit index with bits 8 and 12 forced to 0.
__device__ __forceinline__ unsigned expand_base(unsigned b) {
  return (b & 0x00FFu) | ((b & 0x0700u) << 1) | ((b & 0x3800u) << 2);
}

// ---------------------------------------------------------------------------
// Kernel 1: rotate the coupled |01>/|10> subspace.
//   out01 = c*psi01 - i*s*psi10 ; out10 = c*psi10 - i*s*psi01
// One thread handles 4 batch columns of one (i01,i10) pair: 4x B128 in
// (single address register + immediate row offsets), 16 FMAs (packed f32),
// 4x B128 NT out.
// ---------------------------------------------------------------------------
__global__ __launch_bounds__(256) void iswap_rotate_kernel(
    const float* __restrict__ sr, const float* __restrict__ si,
    const float* __restrict__ tp,
    float* __restrict__ outr, float* __restrict__ outi) {
  unsigned id   = blockIdx.x * 256u + threadIdx.x;  // [0, NBASE*ROWF4)
  unsigned c4   = id & (ROWF4 - 1u);
  unsigned b    = id >> 6;
  unsigned base = expand_base(b);
  unsigned i01  = base | BIT_B;
  unsigned i10  = base | BIT_A;

  float s, c;
  __sincosf(tp[0], &s, &c);

  const v4f* sr4 = (const v4f*)sr;
  const v4f* si4 = (const v4f*)si;
  v4f* or4 = (v4f*)outr;
  v4f* oi4 = (v4f*)outi;

  unsigned o01 = i01 * ROWF4 + c4;
  unsigned o10 = i10 * ROWF4 + c4;

  v4f ar = sr4[o01];  // Re psi01
  v4f ai = si4[o01];  // Im psi01
  v4f br = sr4[o10];  // Re psi10
  v4f bi = si4[o10];  // Im psi10

  v4f r01, m01, r10, m10;
#define ROT1(f)                                \
  r01.f = fmaf(c, ar.f,  s * bi.f);            \
  m01.f = fmaf(c, ai.f, -s * br.f);            \
  r10.f = fmaf(c, br.f,  s * ai.f);            \
  m10.f = fmaf(c, bi.f, -s * ar.f);
  ROT1(x) ROT1(y) ROT1(z) ROT1(w)
#undef ROT1

  __builtin_nontemporal_store(r01, &or4[o01]);
  __builtin_nontemporal_store(m01, &oi4[o01]);
  __builtin_nontemporal_store(r10, &or4[o10]);
  __builtin_nontemporal_store(m10, &oi4[o10]);
}

// ---------------------------------------------------------------------------
// Kernel 2: pass-through copy of the |00> and |11> rows (re + im) using the
// gfx1250 async global<->LDS DMA path (ASYNCcnt). Each lane owns private LDS
// slots, so same-wave s_wait_asynccnt ordering is sufficient (no barrier).
// 4 float4 per lane -> 16 KB staged per block per direction. Stores carry
// TH_STORE_NT so the copy's 64 MB output does not evict the L2-resident input.
// ---------------------------------------------------------------------------
__global__ __launch_bounds__(256) void passthrough_async_kernel(
    const float* __restrict__ sr, const float* __restrict__ si,
    float* __restrict__ outr, float* __restrict__ outi) {
  __shared__ v4f smem[1024];  // 16 KB staging, one slot per (lane, k)
  unsigned tid  = threadIdx.x;
  unsigned lds0 = (unsigned)(uintptr_t)&smem[0];  // low 32 bits = LDS offset

  unsigned offs[4];
  unsigned imag[4];
#pragma unroll
  for (int k = 0; k < 4; ++k) {
    unsigned f   = blockIdx.x * 1024u + (unsigned)k * 256u + tid;
    unsigned c4  = f & (ROWF4 - 1u);
    unsigned rp  = f >> 6;                 // [0, 4*NBASE)
    unsigned b   = rp & (NBASE - 1u);
    unsigned sel = rp >> 14;               // bit0: 00 vs 11, bit1: re vs im
    unsigned row = expand_base(b) | ((sel & 1u) ? (BIT_A | BIT_B) : 0u);
    offs[k] = row * ROWF4 + c4;            // float4 offset within a plane
    imag[k] = sel >> 1;
  }

#pragma unroll
  for (int k = 0; k < 4; ++k) {
    const v4f* src = (const v4f*)(imag[k] ? si : sr) + offs[k];
    unsigned loff  = lds0 + ((unsigned)k * 256u + tid) * 16u;
    asm volatile("global_load_async_to_lds_b128 %0, %1, off"
                 :: "v"(loff), "v"(src) : "memory");
  }

  asm volatile("s_wait_asynccnt 0x0" ::: "memory");

#pragma unroll
  for (int k = 0; k < 4; ++k) {
    v4f* dst      = (v4f*)(imag[k] ? outi : outr) + offs[k];
    unsigned loff = lds0 + ((unsigned)k * 256u + tid) * 16u;
    asm volatile("global_store_async_from_lds_b128 %0, %1, off th:TH_STORE_NT"
                 :: "v"(dst), "v"(loff) : "memory");
  }
  asm volatile("s_wait_asynccnt 0x0" ::: "memory");
}

// ---------------------------------------------------------------------------
extern "C" void kernel_launch(void* const* d_in, const int* in_sizes, int n_in,
                              void* d_out, int out_size, void* d_ws, size_t ws_size,
                              hipStream_t stream) {
  (void)in_sizes; (void)n_in; (void)out_size; (void)d_ws; (void)ws_size;
  const float* sr = (const float*)d_in[0];   // state_real [DIM, BATCH]
  const float* si = (const float*)d_in[1];   // state_imag [DIM, BATCH]
  const float* tp = (const float*)d_in[2];   // t (1 element)
  float* outr = (float*)d_out;                           // out[0] = real
  float* outi = outr + (size_t)DIM_ * BATCH_;            // out[1] = imag

  // Rotate half: NBASE*ROWF4 = 1,048,576 threads -> 4096 blocks x 256.
  iswap_rotate_kernel<<<dim3(4096), dim3(256), 0, stream>>>(sr, si, tp, outr, outi);
  // Copy half: 4,194,304 float4 / 1024 per block -> 4096 blocks x 256.
  passthrough_async_kernel<<<dim3(4096), dim3(256), 0, stream>>>(sr, si, outr, outi);
}